// Nodes_Embedding_87325275062873
// MI455X (gfx1250) — compile-verified
//
#include <hip/hip_runtime.h>

typedef float v2f __attribute__((ext_vector_type(2)));
typedef float v4f __attribute__((ext_vector_type(4)));
typedef float v8f __attribute__((ext_vector_type(8)));

// SWAPX16: exchange lanes 0-15 <-> 16-31 (group-of-32 mode, xor=0x10, and=0x1f)
__device__ __forceinline__ float swap_half(float x) {
    return __int_as_float(__builtin_amdgcn_ds_swizzle(__float_as_int(x), 0x401f));
}

// D = sum_k A_k * B_k + C   (16x16 fp32 GEMM as 4 chained 16x16x4 WMMAs)
__device__ __forceinline__ v8f wmma16x16(const v2f* Af, const v2f* Bf, v8f c) {
#pragma unroll
    for (int kk = 0; kk < 4; ++kk)
        c = __builtin_amdgcn_wmma_f32_16x16x4_f32(
                /*neg_a=*/false, Af[kk], /*neg_b=*/false, Bf[kk],
                /*c_mod=*/(short)0, c, /*reuse_a=*/false, /*reuse_b=*/false);
    return c;
}

__device__ __forceinline__ void elu8(v8f& a) {
#pragma unroll
    for (int v = 0; v < 8; ++v) {
        float x = a[v];
        a[v] = x > 0.0f ? x : (__expf(x) - 1.0f);
    }
}

// Convert D fragment (feat = vgpr + 8*half, atom = lane%16) into the B
// fragment of the next layer (K-chunks of 4 feats, atom = lane%16).
// Only cross-lane movement needed: half-wave swap.
__device__ __forceinline__ void make_bfrag(const v8f& d, v2f* B, int half) {
    float dx[8];
#pragma unroll
    for (int v = 0; v < 8; ++v) dx[v] = swap_half(d[v]);
    const bool hi = (half != 0);
    B[0] = hi ? v2f{dx[2], dx[3]} : v2f{d[0], d[1]};   // feats 0..3
    B[1] = hi ? v2f{dx[6], dx[7]} : v2f{d[4], d[5]};   // feats 4..7
    B[2] = hi ? v2f{d[2], d[3]}   : v2f{dx[0], dx[1]}; // feats 8..11
    B[3] = hi ? v2f{d[6], d[7]}   : v2f{dx[4], dx[5]}; // feats 12..15
}

// A-matrix fragment of W (M=out feat = lane%16, K-chunk kk: K = 4kk + 2*half + j)
__device__ __forceinline__ void load_wfrag(const float* __restrict__ W, v2f* A,
                                           int l16, int half) {
#pragma unroll
    for (int kk = 0; kk < 4; ++kk)
        A[kk] = *reinterpret_cast<const v2f*>(W + l16 * 16 + kk * 4 + half * 2);
}

// C-fragment of bias: c[v] = b[v + 8*half]
__device__ __forceinline__ v8f load_bfrag(const float* __restrict__ b, int half) {
    v4f lo = *reinterpret_cast<const v4f*>(b + half * 8);
    v4f hi = *reinterpret_cast<const v4f*>(b + half * 8 + 4);
    v8f c;
#pragma unroll
    for (int v = 0; v < 4; ++v) { c[v] = lo[v]; c[v + 4] = hi[v]; }
    return c;
}

__global__ __launch_bounds__(256)
void Nodes_Embedding_87325275062873_kernel(
    const float* __restrict__ X,  const float* __restrict__ charge,
    const float* __restrict__ W1, const float* __restrict__ b1,
    const float* __restrict__ W2, const float* __restrict__ b2,
    const float* __restrict__ W3, const float* __restrict__ b3,
    const int*   __restrict__ part, float* __restrict__ out, int ntiles) {
    const int lane = threadIdx.x & 31;
    const int wave = threadIdx.x >> 5;
    const int l16  = lane & 15;
    const int half = lane >> 4;

    // Loop-invariant weight / bias fragments live in registers.
    v2f A1[4], A2[4], A3[4];
    load_wfrag(W1, A1, l16, half);
    load_wfrag(W2, A2, l16, half);
    load_wfrag(W3, A3, l16, half);
    const v8f C1 = load_bfrag(b1, half);
    const v8f C2 = load_bfrag(b2, half);
    const v8f C3 = load_bfrag(b3, half);

    const int wstride = gridDim.x * 8;   // waves in flight
    for (int tile = blockIdx.x * 8 + wave; tile < ntiles; tile += wstride) {
        const int atom = tile * 16 + l16;
        const float* xrow = X + (size_t)atom * 16;

        // B fragment of X^T: per K-chunk, lane loads float2 (coalesced 1KB/tile)
        v2f B[4];
#pragma unroll
        for (int kk = 0; kk < 4; ++kk)
            B[kk] = *reinterpret_cast<const v2f*>(xrow + kk * 4 + half * 2);

        if (tile + wstride < ntiles)  // pull next tile toward the caches
            __builtin_prefetch(X + (size_t)(tile + wstride) * 256 + lane * 8, 0, 0);

        // Layer 1..3: H^T = elu(W * X^T + b)
        v8f h = wmma16x16(A1, B, C1);
        elu8(h);
        v2f Bn[4];
        make_bfrag(h, Bn, half);
        h = wmma16x16(A2, Bn, C2);
        elu8(h);
        make_bfrag(h, Bn, half);
        h = wmma16x16(A3, Bn, C3);
        elu8(h);

        // Per-atom scalar: atom_part + atom_charge[atom_part]
        const int p = part[atom];
        const float add = (float)p + charge[p];
#pragma unroll
        for (int v = 0; v < 8; ++v) h[v] += add;

        // Store: lane owns contiguous 8 floats of its atom's row -> 2x b128
        float* orow = out + (size_t)atom * 16 + half * 8;
        *reinterpret_cast<v4f*>(orow)     = v4f{h[0], h[1], h[2], h[3]};
        *reinterpret_cast<v4f*>(orow + 4) = v4f{h[4], h[5], h[6], h[7]};
    }
}

extern "C" void kernel_launch(void* const* d_in, const int* in_sizes, int n_in,
                              void* d_out, int out_size, void* d_ws, size_t ws_size,
                              hipStream_t stream) {
    const float* X    = (const float*)d_in[0];
    const float* ch   = (const float*)d_in[1];
    const float* W1   = (const float*)d_in[2];
    const float* b1   = (const float*)d_in[3];
    const float* W2   = (const float*)d_in[4];
    const float* b2   = (const float*)d_in[5];
    const float* W3   = (const float*)d_in[6];
    const float* b3   = (const float*)d_in[7];
    const int*   part = (const int*)d_in[8];
    float* out = (float*)d_out;

    const int n_atoms = in_sizes[8];
    const int ntiles  = n_atoms / 16;          // N = 4,000,000 -> 250,000 tiles
    int grid = (ntiles + 7) / 8;               // 8 waves (tiles) per 256-thread block
    if (grid > 2048) grid = 2048;              // grid-stride beyond this
    if (grid < 1) grid = 1;

    hipLaunchKernelGGL(Nodes_Embedding_87325275062873_kernel,
                       dim3(grid), dim3(256), 0, stream,
                       X, ch, W1, b1, W2, b2, W3, b3, part, out, ntiles);
}